// EigenH_86011015069965
// MI455X (gfx1250) — compile-verified
//
#include <hip/hip_runtime.h>
#include <math.h>

typedef float v2f __attribute__((ext_vector_type(2)));
typedef float v8f __attribute__((ext_vector_type(8)));

#define SWEEPS 8

// D = X * Y (16x16 row-major matrices in LDS), via 4 chained f32 WMMAs (K=4 each).
// Wave32; EXEC must be all ones when called.
__device__ __forceinline__ v8f mm16(const float* __restrict__ X,
                                    const float* __restrict__ Y,
                                    bool transX, int lane) {
  const int half = lane >> 4;    // 0 or 1
  const int mn   = lane & 15;    // row (for A-frag) / col (for B-frag)
  v8f acc = {};
#pragma unroll
  for (int jb = 0; jb < 4; ++jb) {
    const int k0 = 4 * jb + 2 * half;
    v2f a, b;
    if (transX) {                         // A-frag of X^T: (m,k) = X[k][m]
      a[0] = X[k0 * 16 + mn];
      a[1] = X[(k0 + 1) * 16 + mn];
    } else {                              // A-frag 16x4: lane m reads X[m][k0], X[m][k0+1]
      a[0] = X[mn * 16 + k0];
      a[1] = X[mn * 16 + k0 + 1];
    }
    b[0] = Y[k0 * 16 + mn];               // B-frag 4x16: lane n reads Y[k0][n], Y[k0+1][n]
    b[1] = Y[(k0 + 1) * 16 + mn];
    acc = __builtin_amdgcn_wmma_f32_16x16x4_f32(false, a, false, b,
                                                (short)0, acc, false, false);
  }
  return acc;
}

// Store C/D fragment (v8f) back to row-major 16x16 LDS tile.
__device__ __forceinline__ void storeCD(float* __restrict__ D, v8f acc, int lane) {
  const int half = lane >> 4, mn = lane & 15;
#pragma unroll
  for (int v = 0; v < 8; ++v) D[(v + 8 * half) * 16 + mn] = acc[v];
}

__global__ __launch_bounds__(32)
void EigenH_kernel(const float* __restrict__ in, float* __restrict__ out, int nmat) {
  __shared__ float S[256], V[256], J[256], T[256];
  __shared__ float esort[16];

  const int g = blockIdx.x;
  if (g >= nmat) return;
  const int lane = threadIdx.x;
  const int half = lane >> 4, mn = lane & 15;

  // ---- Build 16x16 symmetric embedding from packed Hermitian input -------
  // a[b,0,k,t,p]: per-matrix base = g*72; w0 = ab[p], w1 = ab[36+p]
  const float* ab = in + (size_t)g * 72;
  for (int p = lane; p < 36; p += 32) {
    int rr = 0, pp = p;
    while (pp >= 8 - rr) { pp -= 8 - rr; ++rr; }   // triu_indices(8) decode
    int ss = rr + pp;
    float w0 = ab[p];
    float w1 = ab[36 + p];
    int r2 = 2 * rr, s2 = 2 * ss;
    // same write order as reference (.set last-wins for r==s diagonal)
    S[s2 * 16 + r2]             = w0;
    S[r2 * 16 + s2]             = w0;
    S[(s2 + 1) * 16 + (r2 + 1)] = w0;
    S[(r2 + 1) * 16 + (s2 + 1)] = w0;
    S[r2 * 16 + (s2 + 1)]       = -w1;
    S[(s2 + 1) * 16 + r2]       = -w1;
    S[(r2 + 1) * 16 + s2]       = w1;
    S[s2 * 16 + (r2 + 1)]       = w1;
  }
  // V = I
#pragma unroll
  for (int v = 0; v < 8; ++v) {
    int row = v + 8 * half;
    V[row * 16 + mn] = (row == mn) ? 1.f : 0.f;
  }
  __syncthreads();

  // ---- Cyclic Jacobi: 8 sweeps x 15 rounds of 8 disjoint rotations -------
  for (int sweep = 0; sweep < SWEEPS; ++sweep) {
    for (int round = 0; round < 15; ++round) {
      // J = I
#pragma unroll
      for (int v = 0; v < 8; ++v) {
        int row = v + 8 * half;
        J[row * 16 + mn] = (row == mn) ? 1.f : 0.f;
      }
      __syncthreads();
      if (lane < 8) {
        // round-robin tournament pairing over 16 indices
        int p, q;
        if (lane == 0) { p = 15; q = round % 15; }
        else { p = (round + lane) % 15; q = (round - lane + 15) % 15; }
        if (p > q) { int t = p; p = q; q = t; }
        float app = S[p * 16 + p], aqq = S[q * 16 + q], apq = S[p * 16 + q];
        if (fabsf(apq) > 1e-12f * (fabsf(app) + fabsf(aqq)) + 1e-30f) {
          float tau = (aqq - app) / (2.f * apq);
          float t = (tau >= 0.f ? 1.f : -1.f) / (fabsf(tau) + sqrtf(1.f + tau * tau));
          float c = 1.f / sqrtf(1.f + t * t);
          float s = t * c;
          J[p * 16 + p] = c;  J[q * 16 + q] = c;
          J[p * 16 + q] = s;  J[q * 16 + p] = -s;
        }
      }
      __syncthreads();
      // T = J^T * S
      v8f acc = mm16(J, S, true, lane);
      storeCD(T, acc, lane);
      __syncthreads();
      // S = T * J ; V = V * J   (loads of V issue before stores: DS in-order)
      acc = mm16(T, J, false, lane);
      storeCD(S, acc, lane);
      v8f accv = mm16(V, J, false, lane);
      storeCD(V, accv, lane);
      __syncthreads();
    }
  }

  // ---- Sort eigenvalues ascending, permute eigenvector columns -----------
  if (lane < 16) {
    float em = S[lane * 16 + lane];
    int r = 0;
    for (int j2 = 0; j2 < 16; ++j2) {
      float ej = S[j2 * 16 + j2];
      if (ej < em || (ej == em && j2 < lane)) ++r;
    }
    esort[r] = em;
    for (int row = 0; row < 16; ++row)       // T := V with sorted columns
      T[row * 16 + r] = V[row * 16 + lane];
  }
  __syncthreads();

  // ---- Emit outputs: out_v[i,j,c] = Vs[2j][2i+c]; out_d diag(es[2i]) -----
  float* outv = out + (size_t)g * 128;
  float* outd = out + (size_t)nmat * 128 + (size_t)g * 128;
#pragma unroll
  for (int t2 = 0; t2 < 4; ++t2) {
    int idx  = t2 * 32 + lane;     // 0..127 = i*16 + j*2 + comp
    int i    = idx >> 4;
    int j2   = (idx >> 1) & 7;
    int comp = idx & 1;
    outv[idx] = T[(2 * j2) * 16 + (2 * i + comp)];
    outd[idx] = (comp == 0 && i == j2) ? esort[2 * i] : 0.f;
  }
}

extern "C" void kernel_launch(void* const* d_in, const int* in_sizes, int n_in,
                              void* d_out, int out_size, void* d_ws, size_t ws_size,
                              hipStream_t stream) {
  const float* a = (const float*)d_in[0];
  float* out = (float*)d_out;
  const int nmat = in_sizes[0] / 72;   // B*K matrices (72 = 2*P floats each)
  EigenH_kernel<<<nmat, 32, 0, stream>>>(a, out, nmat);
}